// Mask_29119878267365
// MI455X (gfx1250) — compile-verified
//
#include <hip/hip_runtime.h>
#include <stdint.h>

#define THREADS 256
#define BATCH 4  // async B128 transfers per thread per wait

// ---------------------------------------------------------------------------
// Kernel 1: per-row scan for the two MASK_ID positions (exactly 2 per row by
// construction; min = first, max = second, matching ascending order).
// ---------------------------------------------------------------------------
__global__ __launch_bounds__(THREADS)
void find_mask_pos(const int* __restrict__ ids, const int* __restrict__ mask_id_p,
                   int2* __restrict__ pos, int S) {
  __shared__ int smin, smax;
  const int b = blockIdx.x;
  const int mid = *mask_id_p;
  if (threadIdx.x == 0) { smin = 0x7fffffff; smax = -1; }
  __syncthreads();
  for (int s = threadIdx.x; s < S; s += blockDim.x) {
    if (ids[(size_t)b * S + s] == mid) {
      atomicMin(&smin, s);
      atomicMax(&smax, s);
    }
  }
  __syncthreads();
  if (threadIdx.x == 0) pos[b] = make_int2(smin, smax);
}

// ---------------------------------------------------------------------------
// Kernel 2: 256 MB streaming copy via the CDNA5 async global<->LDS DMA path.
// Each lane owns BATCH private 16B LDS slots per buffer; per iteration a wave
// issues BATCH async loads (2 KB), one s_wait_asynccnt 0, BATCH async stores
// (2 KB). Double-buffered: stores from buf[p] drain at the wait inside the
// next iteration, before buf[p] is rewritten in iteration i+2. Non-temporal
// hints keep the 512 MB once-touched stream out of the 192 MB L2. Data never
// touches VGPRs, so register pressure stays minimal and occupancy is maximal.
// ---------------------------------------------------------------------------
__global__ __launch_bounds__(THREADS)
void stream_copy_async(const float4* __restrict__ src, float4* __restrict__ dst,
                       unsigned n4) {
  __shared__ float4 buf[2][BATCH][THREADS];  // 32 KB
  const unsigned tid = threadIdx.x;
  const unsigned stride = gridDim.x * blockDim.x;
  const unsigned span = BATCH * stride;
  unsigned p = 0;
  for (unsigned g = blockIdx.x * blockDim.x + tid; g < n4; g += span, p ^= 1) {
#pragma unroll
    for (unsigned k = 0; k < BATCH; ++k) {
      const unsigned idx = g + k * stride;
      if (idx < n4) {
        const unsigned goff = idx * 16u;  // byte offset; max 2^28 fits in i32
        const unsigned lds_off = (unsigned)(uintptr_t)&buf[p][k][tid];
        // copies 16 bytes from global src + goff to the LDS address in lds_off (async, ASYNCcnt)
        asm volatile("global_load_async_to_lds_b128 %0, %1, %2 th:TH_LOAD_NT"
                     :
                     : "v"(lds_off), "v"(goff), "s"(src)
                     : "memory");
      }
    }
    // Drains this iteration's loads AND the previous iteration's stores.
    asm volatile("s_wait_asynccnt 0" ::: "memory");
#pragma unroll
    for (unsigned k = 0; k < BATCH; ++k) {
      const unsigned idx = g + k * stride;
      if (idx < n4) {
        const unsigned goff = idx * 16u;
        const unsigned lds_off = (unsigned)(uintptr_t)&buf[p][k][tid];
        // copies 16 bytes from the LDS address in lds_off to global dst + goff (async, ASYNCcnt)
        asm volatile("global_store_async_from_lds_b128 %0, %1, %2 th:TH_STORE_NT"
                     :
                     : "v"(goff), "v"(lds_off), "s"(dst)
                     : "memory");
      }
    }
  }
  // s_endpgm performs an implicit wait-idle, draining outstanding stores.
}

// ---------------------------------------------------------------------------
// Kernel 3: overwrite the two mask rows per batch with mask[0], mask[1].
// grid = (B, 2); each block writes one 4 KB row (256 float4).
// ---------------------------------------------------------------------------
__global__ __launch_bounds__(THREADS)
void scatter_mask_rows(const float4* __restrict__ mask, const int2* __restrict__ pos,
                       float4* __restrict__ out, int S, int Dv /* = D/4 */) {
  const int b = blockIdx.x;
  const int which = blockIdx.y;  // 0 -> first mask position, 1 -> second
  const int2 p = pos[b];
  const int row = (which == 0) ? p.x : p.y;
  const float4* srcRow = mask + (size_t)which * Dv;
  float4* dstRow = out + ((size_t)b * S + row) * (size_t)Dv;
  for (int i = threadIdx.x; i < Dv; i += blockDim.x)
    dstRow[i] = srcRow[i];
}

// ---------------------------------------------------------------------------
// Launch: inputs are {input_ids[B*S] i32, input_embed[B*S*D] f32,
// mask[3*D] f32, mask_id[1] i32}; output is [B*S*D] f32.
// Same-stream launches serialize: positions scan, full copy, then scatter.
// ---------------------------------------------------------------------------
extern "C" void kernel_launch(void* const* d_in, const int* in_sizes, int n_in,
                              void* d_out, int out_size, void* d_ws, size_t ws_size,
                              hipStream_t stream) {
  const int*   ids     = (const int*)d_in[0];
  const float* embed   = (const float*)d_in[1];
  const float* mask    = (const float*)d_in[2];
  const int*   mask_id = (const int*)d_in[3];
  float* out = (float*)d_out;
  int2*  pos = (int2*)d_ws;  // B entries, 512 bytes

  const int S = 1024;                    // fixed by the reference
  const int D = in_sizes[2] / 3;         // 1024
  const int B = in_sizes[0] / S;         // 64
  const unsigned n4 = (unsigned)(in_sizes[1] / 4);  // 16,777,216 float4

  find_mask_pos<<<B, THREADS, 0, stream>>>(ids, mask_id, pos, S);
  // 4096 blocks x 8 waves = 32K waves; span = 4096*256*BATCH -> 4 iterations.
  stream_copy_async<<<4096, THREADS, 0, stream>>>(
      (const float4*)embed, (float4*)out, n4);
  scatter_mask_rows<<<dim3(B, 2), THREADS, 0, stream>>>(
      (const float4*)mask, pos, (float4*)out, S, D / 4);
}